// MotionApprFusion_57337813402048
// MI455X (gfx1250) — compile-verified
//
#include <hip/hip_runtime.h>
#include <math.h>

// MI455X / gfx1250, wave32. All matrix math via V_WMMA_F32_16X16X4_F32.
//
// Problem sizes (fixed by reference): B=16, S=2048, D=512, FF=2048, fp32.
//
// Workspace layout (floats), N1 = B*S*D = 16,777,216:
//   [0*N1) qbuf   [1*N1) kbuf   [2*N1) vbuf
//   [3*N1) x0     [4*N1) x1     [5*N1) x2
//   [6*N1) fusion_att
//   [7*N1) scores (S*S = 4,194,304 floats, reused per batch/branch)
//   then fusion_k (B*3*D) and fscore (B*3)
//   hidden (B*S*FF = 4*N1 floats) reuses [0 .. 4*N1) after attention+combine
//   ffn2 out reuses x1.
// Peak ws ~= 490 MB.

#define B_  16
#define S_  2048
#define D_  512
#define FF_ 2048

typedef __attribute__((ext_vector_type(2))) float v2f;
typedef __attribute__((ext_vector_type(8))) float v8f;

// ---------------------------------------------------------------------------
// Generic WMMA f32 GEMM:  C = act( scale * (A @ B) + bias )
//   A: M x K, row-major, leading dim lda
//   B: transB ? (N x K, access B[n*ldb + k])  :  (K x N, access B[k*ldb + n])
//   Block tile 128(M) x 128(N): 8 waves as 4x2, each wave a 32x64 tile held in
//   2x4 v8f accumulators -> 6 fragment loads feed 8 WMMAs per K-chunk.
//   Requires M % 128 == 0, N % 128 == 0, K % 32 == 0 (true for all uses here).
// ---------------------------------------------------------------------------
__global__ __launch_bounds__(256) void gemm_wmma_f32(
    const float* __restrict__ A, const float* __restrict__ Bm,
    const float* __restrict__ bias, float* __restrict__ C,
    int M, int N, int K, int lda, int ldb, int ldc,
    int transB, int relu, float scale)
{
    __shared__ float As[128][33];
    __shared__ float Bs[32][129];

    const int tid  = threadIdx.x;
    const int wave = tid >> 5;
    const int lane = tid & 31;
    const int half = lane >> 4;     // 0 or 1
    const int l16  = lane & 15;
    const int wr   = wave >> 1;     // 0..3  (M sub-tile, 32 rows each)
    const int wc   = wave & 1;      // 0..1  (N sub-tile, 64 cols each)
    const int r0   = blockIdx.y * 128;
    const int c0   = blockIdx.x * 128;
    const int mb   = wr * 32;       // wave row base within block tile
    const int nb   = wc * 64;       // wave col base within block tile

    v8f acc[2][4];
#pragma unroll
    for (int i = 0; i < 2; ++i)
#pragma unroll
        for (int j = 0; j < 4; ++j)
            acc[i][j] = {};

    const int tr = tid >> 3;         // 0..31
    const int tc = (tid & 7) << 2;   // 0,4,...,28

    for (int k0 = 0; k0 < K; k0 += 32) {
        // stage A: 128 x 32 (4 float4 per thread)
        {
#pragma unroll
            for (int rr = 0; rr < 4; ++rr) {
                const int row = tr + rr * 32;
                const float4 a = *(const float4*)(A + (size_t)(r0 + row) * lda + k0 + tc);
                As[row][tc + 0] = a.x; As[row][tc + 1] = a.y;
                As[row][tc + 2] = a.z; As[row][tc + 3] = a.w;
            }
        }
        // stage B: 32(k) x 128(n) (4 float4 per thread)
        if (transB) {
            const int nn  = tid >> 2;         // 0..63 (+64)
            const int kk8 = (tid & 3) << 3;   // 0,8,16,24
#pragma unroll
            for (int cc = 0; cc < 2; ++cc) {
                const int col = nn + cc * 64;
                const float4 b0 = *(const float4*)(Bm + (size_t)(c0 + col) * ldb + k0 + kk8);
                const float4 b1 = *(const float4*)(Bm + (size_t)(c0 + col) * ldb + k0 + kk8 + 4);
                Bs[kk8 + 0][col] = b0.x; Bs[kk8 + 1][col] = b0.y;
                Bs[kk8 + 2][col] = b0.z; Bs[kk8 + 3][col] = b0.w;
                Bs[kk8 + 4][col] = b1.x; Bs[kk8 + 5][col] = b1.y;
                Bs[kk8 + 6][col] = b1.z; Bs[kk8 + 7][col] = b1.w;
            }
        } else {
            const int kk  = tid >> 4;          // 0..15 (+16)
            const int nn4 = (tid & 15) << 2;   // 0..60 (+64)
#pragma unroll
            for (int pp = 0; pp < 2; ++pp) {
#pragma unroll
                for (int cc = 0; cc < 2; ++cc) {
                    const int krow = kk + pp * 16;
                    const int col  = nn4 + cc * 64;
                    const float4 b = *(const float4*)(Bm + (size_t)(k0 + krow) * ldb + c0 + col);
                    Bs[krow][col + 0] = b.x; Bs[krow][col + 1] = b.y;
                    Bs[krow][col + 2] = b.z; Bs[krow][col + 3] = b.w;
                }
            }
        }
        // prefetch next K tile into cache while we compute this one
        if (k0 + 32 < K) {
            __builtin_prefetch(A + (size_t)(r0 + tr) * lda + k0 + 32 + tc, 0, 1);
            if (transB)
                __builtin_prefetch(Bm + (size_t)(c0 + (tid >> 2)) * ldb + k0 + 32 + ((tid & 3) << 3), 0, 1);
            else
                __builtin_prefetch(Bm + (size_t)(k0 + 32 + (tid >> 4)) * ldb + c0 + ((tid & 15) << 2), 0, 1);
        }
        __syncthreads();

#pragma unroll
        for (int kk = 0; kk < 8; ++kk) {
            const int k4 = kk * 4 + 2 * half;   // ISA A-frag: lanes16-31 hold K=2,3
            v2f a0, a1, b0, b1, b2, b3;
            a0.x = As[mb + l16][k4];
            a0.y = As[mb + l16][k4 + 1];
            a1.x = As[mb + 16 + l16][k4];
            a1.y = As[mb + 16 + l16][k4 + 1];
            b0.x = Bs[k4][nb + l16];
            b0.y = Bs[k4 + 1][nb + l16];
            b1.x = Bs[k4][nb + 16 + l16];
            b1.y = Bs[k4 + 1][nb + 16 + l16];
            b2.x = Bs[k4][nb + 32 + l16];
            b2.y = Bs[k4 + 1][nb + 32 + l16];
            b3.x = Bs[k4][nb + 48 + l16];
            b3.y = Bs[k4 + 1][nb + 48 + l16];
            acc[0][0] = __builtin_amdgcn_wmma_f32_16x16x4_f32(false, a0, false, b0, (short)0, acc[0][0], false, false);
            acc[0][1] = __builtin_amdgcn_wmma_f32_16x16x4_f32(false, a0, false, b1, (short)0, acc[0][1], false, false);
            acc[0][2] = __builtin_amdgcn_wmma_f32_16x16x4_f32(false, a0, false, b2, (short)0, acc[0][2], false, false);
            acc[0][3] = __builtin_amdgcn_wmma_f32_16x16x4_f32(false, a0, false, b3, (short)0, acc[0][3], false, false);
            acc[1][0] = __builtin_amdgcn_wmma_f32_16x16x4_f32(false, a1, false, b0, (short)0, acc[1][0], false, false);
            acc[1][1] = __builtin_amdgcn_wmma_f32_16x16x4_f32(false, a1, false, b1, (short)0, acc[1][1], false, false);
            acc[1][2] = __builtin_amdgcn_wmma_f32_16x16x4_f32(false, a1, false, b2, (short)0, acc[1][2], false, false);
            acc[1][3] = __builtin_amdgcn_wmma_f32_16x16x4_f32(false, a1, false, b3, (short)0, acc[1][3], false, false);
        }
        __syncthreads();
    }

    // C/D layout: VGPR i -> row (i + 8*half), col l16
#pragma unroll
    for (int si = 0; si < 2; ++si) {
#pragma unroll
        for (int sj = 0; sj < 4; ++sj) {
            const v8f a = acc[si][sj];
#pragma unroll
            for (int i = 0; i < 8; ++i) {
                const int row = r0 + mb + si * 16 + half * 8 + i;
                const int col = c0 + nb + sj * 16 + l16;
                float v = a[i] * scale;
                if (bias) v += bias[col];
                if (relu) v = fmaxf(v, 0.0f);
                C[(size_t)row * ldc + col] = v;
            }
        }
    }
}

// ---------------------------------------------------------------------------
// Masked softmax over one score row (S_ columns).
// mask(col) = (c0 <= col < c1) * umask[col]; scores already include 1/sqrt(D).
// p = softmax(s - 1e10*(1-m)) * m   (matches reference incl. nan_to_num)
// ---------------------------------------------------------------------------
__global__ __launch_bounds__(256) void softmax_mask_kernel(
    float* __restrict__ scores, const float* __restrict__ umask, int c0, int c1)
{
    const int row = blockIdx.x;
    float* srow = scores + (size_t)row * S_;
    const int tid = threadIdx.x;
    __shared__ float sc[S_];
    __shared__ float red[256];

    float mx = -3.0e38f;
    for (int c = tid; c < S_; c += 256) {
        const float m = (c >= c0 && c < c1) ? umask[c] : 0.0f;
        const float s = srow[c] - 1.0e10f * (1.0f - m);
        sc[c] = s;
        mx = fmaxf(mx, s);
    }
    red[tid] = mx; __syncthreads();
    for (int off = 128; off > 0; off >>= 1) {
        if (tid < off) red[tid] = fmaxf(red[tid], red[tid + off]);
        __syncthreads();
    }
    mx = red[0];
    __syncthreads();

    float sum = 0.0f;
    for (int c = tid; c < S_; c += 256) {
        const float e = __expf(sc[c] - mx);
        sc[c] = e;
        sum += e;
    }
    red[tid] = sum; __syncthreads();
    for (int off = 128; off > 0; off >>= 1) {
        if (tid < off) red[tid] += red[tid + off];
        __syncthreads();
    }
    const float inv = 1.0f / red[0];

    for (int c = tid; c < S_; c += 256) {
        const float m = (c >= c0 && c < c1) ? umask[c] : 0.0f;
        srow[c] = sc[c] * inv * m;
    }
}

// ---------------------------------------------------------------------------
// out[row,:] = LN(res[row,:] + x[row,:]) * g + b      (D_ = 512, 256 threads)
// ---------------------------------------------------------------------------
__global__ __launch_bounds__(256) void residual_ln_kernel(
    const float* __restrict__ res, const float* __restrict__ x,
    const float* __restrict__ g, const float* __restrict__ bta,
    float* __restrict__ out)
{
    const int row = blockIdx.x;
    const int tid = threadIdx.x;
    const float* r  = res + (size_t)row * D_;
    const float* xv = x   + (size_t)row * D_;
    const float v0 = r[tid]       + xv[tid];
    const float v1 = r[tid + 256] + xv[tid + 256];

    __shared__ float red[256];
    red[tid] = v0 + v1; __syncthreads();
    for (int off = 128; off > 0; off >>= 1) {
        if (tid < off) red[tid] += red[tid + off];
        __syncthreads();
    }
    const float mu = red[0] * (1.0f / D_);
    __syncthreads();
    red[tid] = v0 * v0 + v1 * v1; __syncthreads();
    for (int off = 128; off > 0; off >>= 1) {
        if (tid < off) red[tid] += red[tid + off];
        __syncthreads();
    }
    const float var = red[0] * (1.0f / D_) - mu * mu;
    const float inv = rsqrtf(var + 1e-6f);

    out[(size_t)row * D_ + tid]       = g[tid]       * (v0 - mu) * inv + bta[tid];
    out[(size_t)row * D_ + tid + 256] = g[tid + 256] * (v1 - mu) * inv + bta[tid + 256];
}

// out[b*outStride + d] = sum_s x[b,s,d]
__global__ __launch_bounds__(256) void fusion_reduce_kernel(
    const float* __restrict__ x, float* __restrict__ out, int outStride)
{
    const int b   = blockIdx.x;
    const int tid = threadIdx.x;
    float a0 = 0.0f, a1 = 0.0f;
    const float* base = x + (size_t)b * S_ * D_;
    for (int s = 0; s < S_; ++s) {
        a0 += base[(size_t)s * D_ + tid];
        a1 += base[(size_t)s * D_ + tid + 256];
    }
    out[(size_t)b * outStride + tid]       = a0;
    out[(size_t)b * outStride + tid + 256] = a1;
}

// Tiny guide attention: fscore[b, 0..2]
__global__ __launch_bounds__(256) void guide_attn_kernel(
    const float* __restrict__ fk, const float* __restrict__ qh,
    const float* __restrict__ kw, const float* __restrict__ kb,
    const float* __restrict__ qw, const float* __restrict__ qb,
    float* __restrict__ fscore)
{
    const int b   = blockIdx.x;
    const int tid = threadIdx.x;
    __shared__ float sk[3][D_];
    __shared__ float sq[D_];
    for (int i = tid; i < 3 * D_; i += 256) (&sk[0][0])[i] = fk[(size_t)b * 3 * D_ + i];
    for (int i = tid; i < D_; i += 256) sq[i] = qh[(size_t)b * D_ + i];
    __syncthreads();

    float dp0 = 0.0f, dp1 = 0.0f, dp2 = 0.0f;
    for (int d = tid; d < D_; d += 256) {
        float qp = qb[d];
        float k0 = kb[d], k1 = kb[d], k2 = kb[d];
        for (int k = 0; k < D_; ++k) {
            const float wq = qw[(size_t)k * D_ + d];
            const float wk = kw[(size_t)k * D_ + d];
            qp += sq[k] * wq;
            k0 += sk[0][k] * wk;
            k1 += sk[1][k] * wk;
            k2 += sk[2][k] * wk;
        }
        dp0 += qp * k0; dp1 += qp * k1; dp2 += qp * k2;
    }
    __shared__ float red[3][256];
    red[0][tid] = dp0; red[1][tid] = dp1; red[2][tid] = dp2;
    __syncthreads();
    for (int off = 128; off > 0; off >>= 1) {
        if (tid < off) {
            red[0][tid] += red[0][tid + off];
            red[1][tid] += red[1][tid + off];
            red[2][tid] += red[2][tid + off];
        }
        __syncthreads();
    }
    if (tid == 0) {
        const float inv = rsqrtf((float)D_);
        const float s0 = red[0][0] * inv, s1 = red[1][0] * inv, s2 = red[2][0] * inv;
        const float mx = fmaxf(s0, fmaxf(s1, s2));
        const float e0 = __expf(s0 - mx), e1 = __expf(s1 - mx), e2 = __expf(s2 - mx);
        const float isum = 1.0f / (e0 + e1 + e2);
        fscore[b * 3 + 0] = e0 * isum;
        fscore[b * 3 + 1] = e1 * isum;
        fscore[b * 3 + 2] = e2 * isum;
    }
}

// out[idx] = sum_n fscore[b,n] * xn[idx]
__global__ __launch_bounds__(256) void fusion_combine_kernel(
    const float* __restrict__ x0, const float* __restrict__ x1,
    const float* __restrict__ x2, const float* __restrict__ fs,
    float* __restrict__ out)
{
    const size_t idx = (size_t)blockIdx.x * 256 + threadIdx.x;
    const int b = (int)(idx / ((size_t)S_ * D_));
    out[idx] = fs[b * 3 + 0] * x0[idx] + fs[b * 3 + 1] * x1[idx] + fs[b * 3 + 2] * x2[idx];
}

// ---------------------------------------------------------------------------
static inline void launch_gemm(const float* A, const float* Bm, const float* bias,
                               float* C, int M, int N, int K,
                               int lda, int ldb, int ldc,
                               int transB, int relu, float scale, hipStream_t s)
{
    dim3 grid(N / 128, M / 128);
    gemm_wmma_f32<<<grid, dim3(256), 0, s>>>(A, Bm, bias, C, M, N, K,
                                             lda, ldb, ldc, transB, relu, scale);
}

extern "C" void kernel_launch(void* const* d_in, const int* in_sizes, int n_in,
                              void* d_out, int out_size, void* d_ws, size_t ws_size,
                              hipStream_t stream)
{
    (void)in_sizes; (void)n_in; (void)out_size; (void)ws_size;
    const float* U      = (const float*)d_in[0];
    const float* q_hid  = (const float*)d_in[1];
    const float* U_mask = (const float*)d_in[2];

    const float* att_kw[3] = {(const float*)d_in[3],  (const float*)d_in[7],  (const float*)d_in[11]};
    const float* att_kb[3] = {(const float*)d_in[4],  (const float*)d_in[8],  (const float*)d_in[12]};
    const float* att_qw[3] = {(const float*)d_in[5],  (const float*)d_in[9],  (const float*)d_in[13]};
    const float* att_qb[3] = {(const float*)d_in[6],  (const float*)d_in[10], (const float*)d_in[14]};
    const float* gkw = (const float*)d_in[15];
    const float* gkb = (const float*)d_in[16];
    const float* gqw = (const float*)d_in[17];
    const float* gqb = (const float*)d_in[18];
    const float* vw[3] = {(const float*)d_in[19], (const float*)d_in[23], (const float*)d_in[27]};
    const float* vb[3] = {(const float*)d_in[20], (const float*)d_in[24], (const float*)d_in[28]};
    const float* ng[3] = {(const float*)d_in[21], (const float*)d_in[25], (const float*)d_in[29]};
    const float* nbb[3]= {(const float*)d_in[22], (const float*)d_in[26], (const float*)d_in[30]};
    const float* w1 = (const float*)d_in[31];
    const float* b1 = (const float*)d_in[32];
    const float* w2 = (const float*)d_in[33];
    const float* b2 = (const float*)d_in[34];
    const float* normg = (const float*)d_in[35];
    const float* normb = (const float*)d_in[36];

    float* w = (float*)d_ws;
    const size_t N1 = (size_t)B_ * S_ * D_;
    float* qbuf   = w + 0 * N1;
    float* kbuf   = w + 1 * N1;
    float* vbuf   = w + 2 * N1;
    float* xbr[3] = {w + 3 * N1, w + 4 * N1, w + 5 * N1};
    float* fatt   = w + 6 * N1;
    float* scores = w + 7 * N1;                       // S_*S_ floats
    float* fk     = scores + (size_t)S_ * S_;         // B*3*D
    float* fscore = fk + (size_t)B_ * 3 * D_;         // B*3
    float* hidden = w;                                // reuses [0,4*N1) after combine
    float* ffn2   = w + 4 * N1;                       // reuses x1

    const int M = B_ * S_;                            // 32768
    const float iscale = 1.0f / sqrtf((float)D_);
    const int cr0[3] = {0,      S_ / 2, 0 };
    const int cr1[3] = {S_ / 2, S_,     S_};

    for (int br = 0; br < 3; ++br) {
        // projections: Q = U@qw+qb, K = U@kw+kb, V = U@vw+vb
        launch_gemm(U, att_qw[br], att_qb[br], qbuf, M, D_, D_, D_, D_, D_, 0, 0, 1.0f, stream);
        launch_gemm(U, att_kw[br], att_kb[br], kbuf, M, D_, D_, D_, D_, D_, 0, 0, 1.0f, stream);
        launch_gemm(U, vw[br],     vb[br],     vbuf, M, D_, D_, D_, D_, D_, 0, 0, 1.0f, stream);

        for (int b = 0; b < B_; ++b) {
            const size_t off = (size_t)b * S_ * D_;
            // scores = Q @ K^T / sqrt(D)
            launch_gemm(qbuf + off, kbuf + off, nullptr, scores,
                        S_, S_, D_, D_, D_, S_, 1, 0, iscale, stream);
            softmax_mask_kernel<<<S_, 256, 0, stream>>>(scores, U_mask + (size_t)b * S_,
                                                        cr0[br], cr1[br]);
            // x_pre = P @ V
            launch_gemm(scores, vbuf + off, nullptr, xbr[br] + off,
                        S_, D_, S_, S_, D_, D_, 0, 0, 1.0f, stream);
        }
        // x = LN(U + x_pre)   (in-place safe: row read fully before write)
        residual_ln_kernel<<<M, 256, 0, stream>>>(U, xbr[br], ng[br], nbb[br], xbr[br]);
    }

    // fusion_k[b,n,:] = sum_s x_n[b,s,:]
    for (int n = 0; n < 3; ++n)
        fusion_reduce_kernel<<<B_, 256, 0, stream>>>(xbr[n], fk + (size_t)n * D_, 3 * D_);

    guide_attn_kernel<<<B_, 256, 0, stream>>>(fk, q_hid, gkw, gkb, gqw, gqb, fscore);

    fusion_combine_kernel<<<(int)(N1 / 256), 256, 0, stream>>>(xbr[0], xbr[1], xbr[2],
                                                               fscore, fatt);

    // FFN: hidden = relu(fatt@w1+b1); ffn2 = hidden@w2+b2; out = LN(fatt+ffn2)
    launch_gemm(fatt,   w1, b1, hidden, M, FF_, D_,  D_,  FF_, FF_, 0, 1, 1.0f, stream);
    launch_gemm(hidden, w2, b2, ffn2,   M, D_,  FF_, FF_, D_,  D_,  0, 0, 1.0f, stream);
    residual_ln_kernel<<<M, 256, 0, stream>>>(fatt, ffn2, normg, normb, (float*)d_out);
}